// XEmbedding_72808285602169
// MI455X (gfx1250) — compile-verified
//
#include <hip/hip_runtime.h>
#include <hip/hip_bf16.h>
#include <math.h>

#define NN 100000
#define NE 3200000
#define STG 196   // LDS staging row stride (floats), padded to avoid bank conflicts

typedef __attribute__((ext_vector_type(2))) float v2f;
typedef __attribute__((ext_vector_type(8))) float v8f;

// ---- compile-time Frobenius norm^2 of W222 numerator (W_raw + W_raw^T(1,0,2)) ----
constexpr double W222_norm2() {
  double A[5][3][3] = {};
  const double h = 1.9364916731037084425896326998912;  // sqrt(15)/2
  const double q = 1.1180339887498948482045868343656;  // sqrt(5)/2
  A[0][0][2] = h; A[0][2][0] = h;
  A[1][0][1] = h; A[1][1][0] = h;
  A[2][0][0] = -q; A[2][1][1] = 2.0 * q; A[2][2][2] = -q;
  A[3][1][2] = h; A[3][2][1] = h;
  A[4][0][0] = -h; A[4][2][2] = h;
  double n2 = 0.0;
  for (int m = 0; m < 5; ++m)
    for (int n = 0; n < 5; ++n)
      for (int p = 0; p < 5; ++p) {
        double t1 = 0.0, t2 = 0.0;
        for (int i = 0; i < 3; ++i)
          for (int k = 0; k < 3; ++k)
            for (int j = 0; j < 3; ++j) {
              t1 += A[m][i][k] * A[n][k][j] * A[p][j][i];
              t2 += A[n][i][k] * A[m][k][j] * A[p][j][i];
            }
        double w = t1 + t2;
        n2 += w * w;
      }
  return n2;
}
constexpr double W222_N2 = W222_norm2();

__device__ __forceinline__ v8f vz8() {
  v8f z;
  for (int i = 0; i < 8; ++i) z[i] = 0.0f;
  return z;
}

__device__ __forceinline__ v8f wmma4(v2f a, v2f b, v8f c) {
  // V_WMMA_F32_16X16X4_F32 : D = A(16x4) x B(4x16) + C(16x16), full f32 precision
  return __builtin_amdgcn_wmma_f32_16x16x4_f32(false, a, false, b, (short)0, c, false, false);
}

// fast sigmoid: one v_exp_f32 + one v_rcp_f32 (no IEEE div fixup chain)
__device__ __forceinline__ float sigm(float x) {
  return __builtin_amdgcn_rcpf(1.0f + __expf(-x));
}

// -------------------------------------------------------------------------
__global__ void zero_kernel(float* __restrict__ p, int n) {
  int i = blockIdx.x * blockDim.x + threadIdx.x;
  if (i < n) p[i] = 0.0f;
}

// -------------------------------------------------------------------------
__global__ void edge_kernel(const float* __restrict__ pos, const float* __restrict__ x,
                            const int* __restrict__ ei, float* __restrict__ acc,
                            float* __restrict__ erbf, float* __restrict__ ersh) {
  int e = blockIdx.x * blockDim.x + threadIdx.x;
  if (e >= NE) return;
  int src = ei[e];
  int dst = ei[NE + e];
  // pos permuted [1,2,0]
  float ax = pos[3 * src + 1], ay = pos[3 * src + 2], az = pos[3 * src + 0];
  float bx = pos[3 * dst + 1], by = pos[3 * dst + 2], bz = pos[3 * dst + 0];
  float vx = bx - ax, vy = by - ay, vz = bz - az;
  float r = sqrtf(vx * vx + vy * vy + vz * vz);
  float d = r * 0.1f;
  float d2 = d * d, d5 = d2 * d2 * d;
  float fc = (d < 1.0f) ? (1.0f - 21.0f * d5 + 35.0f * d5 * d - 15.0f * d5 * d2) : 0.0f;
  float rinv = __builtin_amdgcn_rcpf(fmaxf(r, 1e-12f));
  float ux = vx * rinv, uy = vy * rinv, uz = vz * rinv;
  const float SQ3 = 1.7320508075688772f;
  const float SQ5 = 2.2360679774997896f;
  const float SQ15 = 3.8729833462074170f;
  float sh[9];
  sh[0] = 1.0f;
  sh[1] = SQ3 * ux; sh[2] = SQ3 * uy; sh[3] = SQ3 * uz;
  sh[4] = SQ15 * ux * uz;
  sh[5] = SQ15 * ux * uy;
  sh[6] = SQ5 * (uy * uy - 0.5f * (ux * ux + uz * uz));
  sh[7] = SQ15 * uy * uz;
  sh[8] = 0.5f * SQ15 * (uz * uz - ux * ux);
  float eb = __expf(-d2);
  float g0 = eb * fc, g1 = d * eb * fc, g2 = d2 * eb * fc;
  float scal[9] = {g0, g1, g1, g1, g2, g2, g2, g2, g2};
#pragma unroll
  for (int c = 0; c < 9; ++c)
    atomicAdd(&acc[dst * 9 + c], x[src * 9 + c] * sh[c] * scal[c]);

  // erbf: sqrt(2/10) * sin(n*pi*r/10) / max(r,eps) * fc,  n = 1..16
  // Chebyshev recurrence: sin((n+1)t) = 2*cos(t)*sin(nt) - sin((n-1)t)
  const float kk = 0.44721359549995793f;        // sqrt(2/10)
  float theta = r * 0.31415926535897931f;       // pi/10
  float s1 = __sinf(theta);
  float c1 = __cosf(theta);
  float twoc = 2.0f * c1;
  float amp = kk * rinv * fc;
  float sp = 0.0f, sn = s1;                     // sin(0), sin(theta)
  float eb16[16];
#pragma unroll
  for (int n = 0; n < 16; ++n) {
    eb16[n] = amp * sn;
    float nx = twoc * sn - sp;
    sp = sn;
    sn = nx;
  }
  float4* eo = (float4*)(erbf + e * 16);        // 64B-aligned: vector stores
  eo[0] = make_float4(eb16[0], eb16[1], eb16[2], eb16[3]);
  eo[1] = make_float4(eb16[4], eb16[5], eb16[6], eb16[7]);
  eo[2] = make_float4(eb16[8], eb16[9], eb16[10], eb16[11]);
  eo[3] = make_float4(eb16[12], eb16[13], eb16[14], eb16[15]);

  // ersh = sph(-vec): l0 same, l1 negated, l2 same
  ersh[e * 9 + 0] = 1.0f;
  ersh[e * 9 + 1] = -sh[1]; ersh[e * 9 + 2] = -sh[2]; ersh[e * 9 + 3] = -sh[3];
  ersh[e * 9 + 4] = sh[4]; ersh[e * 9 + 5] = sh[5]; ersh[e * 9 + 6] = sh[6];
  ersh[e * 9 + 7] = sh[7]; ersh[e * 9 + 8] = sh[8];
}

// -------------------------------------------------------------------------
// Node kernel: 64 threads (2 waves); each wave runs the WMMA pipeline on two
// 16-node tiles. Stats accumulated via wave-level shuffle reduce + atomics.
__global__ __launch_bounds__(64) void node_kernel(
    const float* __restrict__ a, const float* __restrict__ tp,
    const float* __restrict__ W0a, const float* __restrict__ W1a, const float* __restrict__ W2a,
    const float* __restrict__ W0b, const float* __restrict__ W1b, const float* __restrict__ W2b,
    float* __restrict__ h0b, float* __restrict__ h1b, float* __restrict__ h2b,
    float* __restrict__ st) {
  __shared__ float outs[64 * 36];
  __shared__ float stg[2 * 16 * STG];
  const int tid = threadIdx.x;
  const int lane = tid & 31;
  const int wave = tid >> 5;
  const int row = lane & 15;
  const int hi = lane >> 4;   // 0: lanes 0-15, 1: lanes 16-31
  const int koff = hi * 2;    // A/B fragment K-offset (VGPR0 holds K0|K2, VGPR1 K1|K3)
  const int blockBase = blockIdx.x * 64;

  // ---------------- phase 1: per-node tensor products -> LDS ----------------
  {
    int node = blockBase + tid;
    float sv[9];
#pragma unroll
    for (int i = 0; i < 9; ++i) sv[i] = 0.0f;
    if (node < NN) {
#pragma unroll
      for (int i = 0; i < 9; ++i) sv[i] = a[node * 9 + i];
    }
    float tpw[11];
#pragma unroll
    for (int i = 0; i < 11; ++i) tpw[i] = tp[i];
    const float H = 1.9364916731037084f;     // sqrt(15)/2
    const float Q = 1.1180339887498949f;     // sqrt(5)/2
    const float ISQ3 = 0.57735026918962584f; // 1/sqrt(3)
    const float ISQ5 = 0.44721359549995793f; // 1/sqrt(5)
    const float CINV = 0.16329931618554518f; // 1/sqrt(37.5) (C112 normalizer)
    float s0 = sv[0], X = sv[1], Y = sv[2], Z = sv[3];
    float t0 = sv[4], t1 = sv[5], t2 = sv[6], t3 = sv[7], t4 = sv[8];
    float* po = &outs[tid * 36];
    // out0
    po[0] = tpw[0] * s0 * s0;
    po[1] = tpw[1] * (X * X + Y * Y + Z * Z) * ISQ3;
    po[2] = tpw[2] * (t0 * t0 + t1 * t1 + t2 * t2 + t3 * t3 + t4 * t4) * ISQ5;
    // M = sum_m s2_m * A_m   (symmetric 3x3)
    float m00 = -Q * t2 - H * t4, m01 = H * t1, m02 = H * t0;
    float m11 = 2.0f * Q * t2, m12 = H * t3, m22 = -Q * t2 + H * t4;
    float r0 = m00 * X + m01 * Y + m02 * Z;
    float r1 = m01 * X + m11 * Y + m12 * Z;
    float r2 = m02 * X + m12 * Y + m22 * Z;
    // out1 : [3 + u*3 + m]
    po[3] = tpw[3] * s0 * X;  po[4] = tpw[3] * s0 * Y;  po[5] = tpw[3] * s0 * Z;
    po[6] = tpw[4] * s0 * X;  po[7] = tpw[4] * s0 * Y;  po[8] = tpw[4] * s0 * Z;
    po[9] = tpw[5] * CINV * r0;  po[10] = tpw[5] * CINV * r1;  po[11] = tpw[5] * CINV * r2;
    po[12] = tpw[6] * CINV * r0; po[13] = tpw[6] * CINV * r1;  po[14] = tpw[6] * CINV * r2;
    // out2 : [15 + u*5 + m]
    po[15] = tpw[7] * s0 * t0; po[16] = tpw[7] * s0 * t1; po[17] = tpw[7] * s0 * t2;
    po[18] = tpw[7] * s0 * t3; po[19] = tpw[7] * s0 * t4;
    po[20] = tpw[8] * s0 * t0; po[21] = tpw[8] * s0 * t1; po[22] = tpw[8] * s0 * t2;
    po[23] = tpw[8] * s0 * t3; po[24] = tpw[8] * s0 * t4;
    // s1^T A_m s1 (= sh2 of s1)
    float v0 = 2.0f * H * X * Z, v1 = 2.0f * H * X * Y;
    float v2 = Q * (-X * X + 2.0f * Y * Y - Z * Z);
    float v3 = 2.0f * H * Y * Z, v4 = H * (Z * Z - X * X);
    po[25] = tpw[9] * CINV * v0; po[26] = tpw[9] * CINV * v1; po[27] = tpw[9] * CINV * v2;
    po[28] = tpw[9] * CINV * v3; po[29] = tpw[9] * CINV * v4;
    // W222 row: 2*trace(M*M*A_c)/||W||
    float a00 = m00 * m00 + m01 * m01 + m02 * m02;
    float a01 = m00 * m01 + m01 * m11 + m02 * m12;
    float a02 = m00 * m02 + m01 * m12 + m02 * m22;
    float a11 = m01 * m01 + m11 * m11 + m12 * m12;
    float a12 = m01 * m02 + m11 * m12 + m12 * m22;
    float a22 = m02 * m02 + m12 * m12 + m22 * m22;
    const float WSC = 2.0f / sqrtf((float)W222_N2);  // compile-time constant
    float w0 = 2.0f * H * a02, w1 = 2.0f * H * a01;
    float w2 = Q * (-a00 + 2.0f * a11 - a22);
    float w3 = 2.0f * H * a12, w4 = H * (a22 - a00);
    po[30] = tpw[10] * WSC * w0; po[31] = tpw[10] * WSC * w1; po[32] = tpw[10] * WSC * w2;
    po[33] = tpw[10] * WSC * w3; po[34] = tpw[10] * WSC * w4;
    po[35] = 0.0f;
  }
  __syncthreads();

  float* ws = &stg[wave * 16 * STG];

  for (int t = 0; t < 2; ++t) {
    const int tLocal = wave * 32 + t * 16;     // tile base within block
    const int tileNode = blockBase + tLocal;   // global node base of tile
    const float* po = &outs[tLocal * 36];

    // ================= h0 : sigmoid(out0 @ W0a) @ W0b =================
    for (int j = 0; j < 8; ++j) {
      int col = j * 16 + row;
      v2f av, bv;
      av[0] = (hi == 0) ? po[row * 36 + 0] : po[row * 36 + 2];
      av[1] = (hi == 0) ? po[row * 36 + 1] : 0.0f;
      bv[0] = (hi == 0) ? W0a[0 * 128 + col] : W0a[2 * 128 + col];
      bv[1] = (hi == 0) ? W0a[1 * 128 + col] : 0.0f;
      v8f c = wmma4(av, bv, vz8());
#pragma unroll
      for (int g = 0; g < 8; ++g)
        ws[(g + hi * 8) * STG + col] = sigm(c[g]);
    }
    __syncthreads();
    for (int j = 0; j < 8; ++j) {
      int col = j * 16 + row;
      v8f acc = vz8();
      for (int kt = 0; kt < 32; ++kt) {
        v2f av, bv;
        av[0] = ws[row * STG + 4 * kt + koff];
        av[1] = ws[row * STG + 4 * kt + koff + 1];
        bv[0] = W0b[(4 * kt + koff) * 128 + col];
        bv[1] = W0b[(4 * kt + koff + 1) * 128 + col];
        acc = wmma4(av, bv, acc);
      }
      float psum = 0.0f, psq = 0.0f;
#pragma unroll
      for (int g = 0; g < 8; ++g) {
        int nd = tileNode + g + hi * 8;
        float v = acc[g];
        if (nd < NN) { h0b[nd * 128 + col] = v; psum += v; psq += v * v; }
      }
      psum += __shfl_xor(psum, 16, 32);
      psq += __shfl_xor(psq, 16, 32);
      if (hi == 0) { atomicAdd(&st[col], psum); atomicAdd(&st[128 + col], psq); }
    }
    __syncthreads();

    // ================= h1 : gate(out1 @ W1a) @ W1b (m = 0..2) =================
    for (int j = 0; j < 4; ++j) {
      int col = j * 16 + row;
      v8f c1[3];
#pragma unroll
      for (int m = 0; m < 3; ++m) {
        v2f av, bv;
        av[0] = po[row * 36 + 3 + (koff + 0) * 3 + m];
        av[1] = po[row * 36 + 3 + (koff + 1) * 3 + m];
        bv[0] = W1a[(koff + 0) * 64 + col];
        bv[1] = W1a[(koff + 1) * 64 + col];
        c1[m] = wmma4(av, bv, vz8());
      }
#pragma unroll
      for (int g = 0; g < 8; ++g) {
        float n2 = c1[0][g] * c1[0][g] + c1[1][g] * c1[1][g] + c1[2][g] * c1[2][g];
        float gate = sigm(sqrtf(n2));
        int nl = g + hi * 8;
        ws[nl * STG + 0 * 64 + col] = c1[0][g] * gate;
        ws[nl * STG + 1 * 64 + col] = c1[1][g] * gate;
        ws[nl * STG + 2 * 64 + col] = c1[2][g] * gate;
      }
    }
    __syncthreads();
    for (int j2 = 0; j2 < 4; ++j2) {
      int col = j2 * 16 + row;
      float ssp = 0.0f;
      for (int m = 0; m < 3; ++m) {
        v8f acc = vz8();
        for (int kt = 0; kt < 16; ++kt) {
          v2f av, bv;
          av[0] = ws[row * STG + m * 64 + 4 * kt + koff];
          av[1] = ws[row * STG + m * 64 + 4 * kt + koff + 1];
          bv[0] = W1b[(4 * kt + koff) * 64 + col];
          bv[1] = W1b[(4 * kt + koff + 1) * 64 + col];
          acc = wmma4(av, bv, acc);
        }
#pragma unroll
        for (int g = 0; g < 8; ++g) {
          int nd = tileNode + g + hi * 8;
          float v = acc[g];
          if (nd < NN) { h1b[nd * 192 + col * 3 + m] = v; ssp += v * v; }
        }
      }
      ssp += __shfl_xor(ssp, 16, 32);
      if (hi == 0) atomicAdd(&st[256 + col], ssp);
    }
    __syncthreads();

    // ================= h2 : gate(out2 @ W2a) @ W2b (m = 0..4) =================
    for (int j = 0; j < 2; ++j) {
      int col = j * 16 + row;
      v8f c2[5];
#pragma unroll
      for (int m = 0; m < 5; ++m) {
        v2f av, bv;
        av[0] = po[row * 36 + 15 + (koff + 0) * 5 + m];
        av[1] = po[row * 36 + 15 + (koff + 1) * 5 + m];
        bv[0] = W2a[(koff + 0) * 32 + col];
        bv[1] = W2a[(koff + 1) * 32 + col];
        c2[m] = wmma4(av, bv, vz8());
      }
#pragma unroll
      for (int g = 0; g < 8; ++g) {
        float n2 = 0.0f;
#pragma unroll
        for (int m = 0; m < 5; ++m) n2 += c2[m][g] * c2[m][g];
        float gate = sigm(sqrtf(n2));
        int nl = g + hi * 8;
#pragma unroll
        for (int m = 0; m < 5; ++m) ws[nl * STG + m * 32 + col] = c2[m][g] * gate;
      }
    }
    __syncthreads();
    for (int j2 = 0; j2 < 2; ++j2) {
      int col = j2 * 16 + row;
      float ssp = 0.0f;
      for (int m = 0; m < 5; ++m) {
        v8f acc = vz8();
        for (int kt = 0; kt < 8; ++kt) {
          v2f av, bv;
          av[0] = ws[row * STG + m * 32 + 4 * kt + koff];
          av[1] = ws[row * STG + m * 32 + 4 * kt + koff + 1];
          bv[0] = W2b[(4 * kt + koff) * 32 + col];
          bv[1] = W2b[(4 * kt + koff + 1) * 32 + col];
          acc = wmma4(av, bv, acc);
        }
#pragma unroll
        for (int g = 0; g < 8; ++g) {
          int nd = tileNode + g + hi * 8;
          float v = acc[g];
          if (nd < NN) { h2b[nd * 160 + col * 5 + m] = v; ssp += v * v; }
        }
      }
      ssp += __shfl_xor(ssp, 16, 32);
      if (hi == 0) atomicAdd(&st[320 + col], ssp);
    }
    __syncthreads();
  }
}

// -------------------------------------------------------------------------
__global__ void finalize_kernel(float* __restrict__ st) {
  int t = threadIdx.x;
  const float invN = 1.0f / (float)NN;
  if (t < 128) {
    float mean = st[t] * invN;
    float var = st[128 + t] * invN - mean * mean;
    st[384 + t] = mean;
    st[512 + t] = rsqrtf(var + 1e-5f);
  }
  if (t < 64) st[640 + t] = rsqrtf(st[256 + t] * invN * (1.0f / 3.0f) + 1e-5f);
  if (t < 32) st[704 + t] = rsqrtf(st[320 + t] * invN * (1.0f / 5.0f) + 1e-5f);
}

// -------------------------------------------------------------------------
__global__ void apply_kernel(const float* __restrict__ h0b, const float* __restrict__ h1b,
                             const float* __restrict__ h2b, const float* __restrict__ st,
                             float* __restrict__ out) {
  int n = blockIdx.x;
  int k = threadIdx.x;
  float v;
  if (k < 128) {
    v = (h0b[n * 128 + k] - st[384 + k]) * st[512 + k];
  } else if (k < 320) {
    int t = k - 128;
    v = h1b[n * 192 + t] * st[640 + t / 3];
  } else {
    int t = k - 320;
    v = h2b[n * 160 + t] * st[704 + t / 5];
  }
  out[n * 480 + k] = v;
}

// -------------------------------------------------------------------------
extern "C" void kernel_launch(void* const* d_in, const int* in_sizes, int n_in,
                              void* d_out, int out_size, void* d_ws, size_t ws_size,
                              hipStream_t stream) {
  (void)in_sizes; (void)n_in; (void)out_size; (void)ws_size;
  const float* x   = (const float*)d_in[0];
  const float* pos = (const float*)d_in[1];
  const int*   ei  = (const int*)d_in[2];
  const float* tpw = (const float*)d_in[3];
  const float* W0a = (const float*)d_in[4];
  const float* W1a = (const float*)d_in[5];
  const float* W2a = (const float*)d_in[6];
  const float* W0b = (const float*)d_in[7];
  const float* W1b = (const float*)d_in[8];
  const float* W2b = (const float*)d_in[9];
  float* out = (float*)d_out;
  float* ws  = (float*)d_ws;

  float* a_acc = ws;                 // 900,000 floats   (node accumulators)
  float* stats = ws + 900000;        // 736 floats       (sums + derived scales)
  float* h0b   = ws + 901120;        // 12,800,000
  float* h1b   = h0b + 12800000;     // 19,200,000
  float* h2b   = h1b + 19200000;     // 16,000,000
  float* erbf  = out + 48000000;     // 3.2M x 16
  float* ersh  = out + 99200000;     // 3.2M x 9

  zero_kernel<<<(900736 + 255) / 256, 256, 0, stream>>>(ws, 900736);
  edge_kernel<<<(NE + 255) / 256, 256, 0, stream>>>(pos, x, ei, a_acc, erbf, ersh);
  node_kernel<<<(NN + 63) / 64, 64, 0, stream>>>(a_acc, tpw, W0a, W1a, W2a,
                                                 W0b, W1b, W2b, h0b, h1b, h2b, stats);
  finalize_kernel<<<1, 128, 0, stream>>>(stats);
  apply_kernel<<<NN, 480, 0, stream>>>(h0b, h1b, h2b, stats, out);
}